// M_11879879543048
// MI455X (gfx1250) — compile-verified
//
#include <hip/hip_runtime.h>
#include <stdint.h>

typedef unsigned int u32;

#define HIST_SIZE 8192      // 13-bit buckets over order-preserving key
#define KEY_SHIFT 19
#define CAND_MAX  8192      // candidate cap (expected ~2-3K for this data)
#define HB 512              // histogram block size
#define CB 256              // compact block size
#define SB 1024             // sort/pick block size
#define NP_MAX 512          // max partial histograms (ws-dependent)

// ---- order-preserving float -> uint key (handles negatives too) ----
__device__ __forceinline__ u32 fkey(float s) {
    u32 u = __float_as_uint(s);
    u ^= ((int)u < 0) ? 0xFFFFFFFFu : 0x80000000u;
    return u;
}

// ---- CDNA5 async global->LDS load, 16B per lane (ASYNCcnt-tracked) ----
__device__ __forceinline__ void async_ld16(const void* gaddr, u32 lds_off) {
    asm volatile("global_load_async_to_lds_b128 %0, %1, off"
                 :: "v"(lds_off), "v"(gaddr) : "memory");
}
__device__ __forceinline__ void wait_async0() {
    asm volatile("s_wait_asynccnt 0" ::: "memory");
}
__device__ __forceinline__ void wait_async1() {
    asm volatile("s_wait_asynccnt 1" ::: "memory");
}

// ---------------- kernel 0: zero counters ----------------
__global__ void k_init(u32* ctrs) {
    int i = blockIdx.x * blockDim.x + threadIdx.x;
    if (i < 16) ctrs[i] = 0u;
}

// ---------------- kernel 1: per-block private histograms (atomic-free flush) ----
// Streams scores through LDS via double-buffered async-to-LDS DMA; each block
// writes its full 8192-entry histogram to part[blockIdx] with plain stores.
__global__ __launch_bounds__(HB) void k_hist(const float* __restrict__ scores,
                                             const float* __restrict__ thr_p,
                                             u32* __restrict__ part,
                                             long long n) {
    __shared__ u32 lhist[HIST_SIZE];          // 32 KB
    __shared__ float4 stage[2][HB];           // 16 KB staging
    const int tid = threadIdx.x;

    for (int i = tid; i < HIST_SIZE; i += HB) lhist[i] = 0u;
    __syncthreads();

    const float thr = *thr_p;
    const float4* __restrict__ src = (const float4*)scores;
    const long long nv = n >> 2;
    const long long stride = (long long)gridDim.x * HB;
    const long long base = (long long)blockIdx.x * HB + tid;
    const u32 lds0 = (u32)(size_t)&stage[0][tid];
    const u32 lds1 = (u32)(size_t)&stage[1][tid];

    const long long iters = (nv + stride - 1) / stride;   // uniform trip count

    {   // prologue: issue first chunk (clamped address keeps tail lanes safe)
        long long i0 = base < nv ? base : (nv - 1);
        async_ld16((const void*)(src + i0), lds0);
    }
    int buf = 0;
    for (long long t = 0; t < iters; ++t) {
        const long long cur = base + t * stride;
        if (t + 1 < iters) {                                // uniform branch
            long long nxt = cur + stride;
            long long in = nxt < nv ? nxt : (nv - 1);
            async_ld16((const void*)(src + in), buf ? lds0 : lds1);
            wait_async1();                                  // drain current, keep next in flight
        } else {
            wait_async0();
        }
        float4 v = stage[buf][tid];                         // ds_load_b128
        if (cur < nv) {
            float e[4] = {v.x, v.y, v.z, v.w};
#pragma unroll
            for (int c = 0; c < 4; ++c) {
                float s = e[c];
                if (s > thr) atomicAdd(&lhist[fkey(s) >> KEY_SHIFT], 1u);
            }
        }
        buf ^= 1;
    }
    __syncthreads();

    u32* __restrict__ my = part + (size_t)blockIdx.x * HIST_SIZE;
    for (int i = tid; i < HIST_SIZE; i += HB) my[i] = lhist[i];   // coalesced, no atomics
}

// ---------------- kernel 1b: column-sum the partial histograms ----------------
__global__ __launch_bounds__(256) void k_reduce(const u32* __restrict__ part,
                                                u32* __restrict__ hist, int np) {
    int b = blockIdx.x * 256 + threadIdx.x;   // one thread per bucket
    u32 s = 0;
    for (int p = 0; p < np; ++p) s += part[(size_t)p * HIST_SIZE + b];
    hist[b] = s;
}

// ---------------- kernel 2: find threshold bucket B (single block) ----------------
// ctrs[1] = B (candidates are elements with bucket >= B), ctrs[2] = count strictly above B
__global__ __launch_bounds__(SB) void k_pick(const u32* __restrict__ hist,
                                             u32* __restrict__ ctrs, int K) {
    __shared__ u32 lh[HIST_SIZE];
    __shared__ u32 chunk[SB];
    const int tid = threadIdx.x;

    for (int i = tid; i < HIST_SIZE; i += SB) lh[i] = hist[i];
    __syncthreads();

    u32 s = 0;
#pragma unroll
    for (int j = 0; j < HIST_SIZE / SB; ++j) s += lh[tid * (HIST_SIZE / SB) + j];
    chunk[tid] = s;
    __syncthreads();

    if (tid == 0) {
        const int CW = HIST_SIZE / SB;   // buckets per chunk
        u32 acc = 0;
        for (int c = SB - 1; c >= 0; --c) {
            if (acc + chunk[c] >= (u32)K) {
                for (int b = c * CW + CW - 1; b >= c * CW; --b) {
                    if (acc + lh[b] >= (u32)K) { ctrs[1] = (u32)b; ctrs[2] = acc; return; }
                    acc += lh[b];
                }
            }
            acc += chunk[c];
        }
        ctrs[1] = 0u;   // fewer than K valid: take everything
        ctrs[2] = acc;
    }
}

// ---------------- kernel 3: compact candidates ----------------
__global__ __launch_bounds__(CB) void k_compact(const float* __restrict__ scores,
                                                const float* __restrict__ thr_p,
                                                u32* __restrict__ ctrs,
                                                float* __restrict__ cval,
                                                u32* __restrict__ cidx,
                                                long long n) {
    const float thr = *thr_p;
    const u32 B = ctrs[1];
    const float4* __restrict__ src = (const float4*)scores;
    const long long nv = n >> 2;
    const long long stride = (long long)gridDim.x * CB;
    for (long long i = (long long)blockIdx.x * CB + threadIdx.x; i < nv; i += stride) {
        float4 v = src[i];
        float e[4] = {v.x, v.y, v.z, v.w};
#pragma unroll
        for (int c = 0; c < 4; ++c) {
            float s = e[c];
            if (s > thr && (fkey(s) >> KEY_SHIFT) >= B) {
                u32 pos = atomicAdd(&ctrs[0], 1u);
                if (pos < CAND_MAX) {
                    cval[pos] = s;
                    cidx[pos] = (u32)(i * 4 + c);
                }
            }
        }
    }
}

// ---------------- kernel 4: bitonic sort candidates + write outputs ----------------
__global__ __launch_bounds__(SB) void k_sort_out(const float* __restrict__ cval,
                                                 const u32* __restrict__ cidx,
                                                 const u32* __restrict__ ctrs,
                                                 const float* __restrict__ results,
                                                 float* __restrict__ out,
                                                 int K, int C) {
    __shared__ float sv[CAND_MAX];   // 32 KB
    __shared__ u32  si[CAND_MAX];    // 32 KB
    const int tid = threadIdx.x;
    u32 nc = ctrs[0];
    if (nc > CAND_MAX) nc = CAND_MAX;

    // sort size: next pow2 covering both candidates and K (uniform across block)
    u32 m = 1;
    while (m < nc || m < (u32)K) m <<= 1;
    if (m > CAND_MAX) m = CAND_MAX;

    for (u32 i = tid; i < m; i += SB) {
        if (i < nc) { sv[i] = cval[i]; si[i] = cidx[i]; }
        else        { sv[i] = -__builtin_inff(); si[i] = 0xFFFFFFFFu; }
    }
    __syncthreads();

    // bitonic sort over m elements, descending by (value desc, index asc)
    for (u32 k = 2; k <= m; k <<= 1) {
        for (u32 j = k >> 1; j > 0; j >>= 1) {
            for (u32 i = tid; i < m; i += SB) {
                u32 ixj = i ^ j;
                if (ixj > i) {
                    float a = sv[i], b = sv[ixj];
                    u32 ia = si[i], ib = si[ixj];
                    bool desc = ((i & k) == 0);
                    bool aGTb = (a > b) || (a == b && ia < ib);
                    bool bGTa = (b > a) || (b == a && ib < ia);
                    bool sw = desc ? bGTa : aGTb;
                    if (sw) { sv[i] = b; sv[ixj] = a; si[i] = ib; si[ixj] = ia; }
                }
            }
            __syncthreads();
        }
    }

    // outputs: [scores K][labels K][keep_idxs K][results K*4]
    for (int i = tid; i < K; i += SB) {
        float s = sv[i];
        u32 f = si[i];
        u32 keep = 0, lab = 0;
        bool valid = (f != 0xFFFFFFFFu);
        if (valid) { keep = f / (u32)C; lab = f % (u32)C; }
        else       { s = -__builtin_inff(); }
        out[i]         = s;
        out[K + i]     = (float)lab;
        out[2 * K + i] = (float)keep;
        const float* r = results + (size_t)keep * 4u;
#pragma unroll
        for (int j = 0; j < 4; ++j)
            out[3 * K + i * 4 + j] = valid ? r[j] : 0.0f;
    }
}

extern "C" void kernel_launch(void* const* d_in, const int* in_sizes, int n_in,
                              void* d_out, int out_size, void* d_ws, size_t ws_size,
                              hipStream_t stream) {
    const float* scores  = (const float*)d_in[0];
    const float* thr_p   = (const float*)d_in[1];
    const float* results = (const float*)d_in[2];
    const long long n = (long long)in_sizes[0];          // 40,000,000
    const int rows = in_sizes[2] / 4;                    // 500,000
    const int C = (int)(n / rows);                       // 80
    const int K = out_size / 7;                          // 1000

    // workspace layout: [hist 32KB][ctrs 64B][cval 32KB][cidx 32KB][partials NP*32KB]
    uint8_t* ws = (uint8_t*)d_ws;
    u32*   hist = (u32*)(ws);
    u32*   ctrs = (u32*)(ws + HIST_SIZE * sizeof(u32));
    float* cval = (float*)(ws + HIST_SIZE * sizeof(u32) + 64);
    u32*   cidx = (u32*)(ws + HIST_SIZE * sizeof(u32) + 64 + CAND_MAX * sizeof(float));
    const size_t fixed = HIST_SIZE * sizeof(u32) + 64 + CAND_MAX * 8;
    u32*   part = (u32*)(ws + fixed);

    // number of partial histograms limited by available scratch
    int np = NP_MAX;
    if (ws_size > fixed) {
        size_t avail = (ws_size - fixed) / (HIST_SIZE * sizeof(u32));
        if ((size_t)np > avail) np = (int)avail;
    } else {
        np = 1;
    }
    if (np < 1) np = 1;

    k_init<<<1, 64, 0, stream>>>(ctrs);
    k_hist<<<np, HB, 0, stream>>>(scores, thr_p, part, n);
    k_reduce<<<HIST_SIZE / 256, 256, 0, stream>>>(part, hist, np);
    k_pick<<<1, SB, 0, stream>>>(hist, ctrs, K);
    k_compact<<<2048, CB, 0, stream>>>(scores, thr_p, ctrs, cval, cidx, n);
    k_sort_out<<<1, SB, 0, stream>>>(cval, cidx, ctrs, results, (float*)d_out, K, C);
}